// QNN_IBM_25125558681860
// MI455X (gfx1250) — compile-verified
//
#include <hip/hip_runtime.h>
#include <math.h>

#define NQ 8
#define DIM 256
#define NELEM (DIM * DIM)          // 65536 complex entries per rho
#define LAYERS 2
#define BATCH 64
#define NSOPS 13                   // batch-independent pair superoperators

typedef float v2f __attribute__((ext_vector_type(2)));
typedef float v8f __attribute__((ext_vector_type(8)));

// ---------------- complex helpers ----------------
struct cplx { float re, im; };
__device__ inline cplx cmul(cplx a, cplx b) { return {a.re * b.re - a.im * b.im, a.re * b.im + a.im * b.re}; }
__device__ inline cplx cadd(cplx a, cplx b) { return {a.re + b.re, a.im + b.im}; }
__device__ inline cplx conjc(cplx a) { return {a.re, -a.im}; }
__device__ inline cplx cis(float a) { float s, c; __sincosf(a, &s, &c); return {c, s}; }

// Kraus -> 4x4 superoperator: S[(p,q),(i,j)] = sum_k K[k][p][i] * conj(K[k][q][j])
__device__ void super_from_kraus(const cplx K[][2][2], int nk, cplx S[4][4]) {
  for (int p = 0; p < 2; ++p)
    for (int q = 0; q < 2; ++q)
      for (int i = 0; i < 2; ++i)
        for (int j = 0; j < 2; ++j) {
          cplx acc = {0.f, 0.f};
          for (int k = 0; k < nk; ++k)
            acc = cadd(acc, cmul(K[k][p][i], conjc(K[k][q][j])));
          S[p * 2 + q][i * 2 + j] = acc;
        }
}

__device__ void mat4_lmul(const cplx A[4][4], cplx S[4][4]) {  // S = A * S
  cplx T[4][4];
  for (int m = 0; m < 4; ++m)
    for (int k = 0; k < 4; ++k) {
      cplx acc = {0.f, 0.f};
      for (int t = 0; t < 4; ++t) acc = cadd(acc, cmul(A[m][t], S[t][k]));
      T[m][k] = acc;
    }
  for (int m = 0; m < 4; ++m)
    for (int k = 0; k < 4; ++k) S[m][k] = T[m][k];
}

// ---------------- noise channel superoperators (IBM Kingston constants) ----------------
#define EPC_1Q 0.0002335f
#define EPC_2Q 0.002137f
#define T1C    0.00022438f
#define T2C    0.00012053f
#define T_RX   3.5e-08f
#define T_RZZ  2.5e-07f

__device__ void dep_super(float p, cplx S[4][4]) {
  cplx K[4][2][2] = {};
  float a = sqrtf(1.f - p), b = sqrtf(p / 3.f);
  K[0][0][0] = {a, 0}; K[0][1][1] = {a, 0};                 // a*I
  K[1][0][1] = {b, 0}; K[1][1][0] = {b, 0};                 // b*X
  K[2][0][1] = {0, -b}; K[2][1][0] = {0, b};                // b*Y
  K[3][0][0] = {b, 0}; K[3][1][1] = {-b, 0};                // b*Z
  super_from_kraus(K, 4, S);
}
__device__ void ad_super(float g, cplx S[4][4]) {
  cplx K[2][2][2] = {};
  K[0][0][0] = {1.f, 0}; K[0][1][1] = {sqrtf(1.f - g), 0};
  K[1][0][1] = {sqrtf(g), 0};
  super_from_kraus(K, 2, S);
}
__device__ void pd_super(float g, cplx S[4][4]) {
  cplx K[2][2][2] = {};
  K[0][0][0] = {1.f, 0}; K[0][1][1] = {sqrtf(1.f - g), 0};
  K[1][1][1] = {sqrtf(g), 0};
  super_from_kraus(K, 2, S);
}
__device__ void rxu_super(float th, cplx S[4][4]) {
  cplx K[1][2][2];
  float c = cosf(0.5f * th), s = sinf(0.5f * th);
  K[0][0][0] = {c, 0}; K[0][0][1] = {0, -s};
  K[0][1][0] = {0, -s}; K[0][1][1] = {c, 0};
  super_from_kraus(K, 1, S);
}
// left-compose rz(phi) diagonal superop: row m=(i,j) scaled by exp(i*phi/2*(zj-zi))
__device__ void rz_lmul(float phi, cplx S[4][4]) {
  for (int i = 0; i < 2; ++i)
    for (int j = 0; j < 2; ++j) {
      float zi = 1.f - 2.f * i, zj = 1.f - 2.f * j;
      cplx d = cis(0.5f * phi * (zj - zi));
      int m = i * 2 + j;
      for (int k = 0; k < 4; ++k) S[m][k] = cmul(d, S[m][k]);
    }
}

// noisy_rx superop: PD * AD * DEP1 * RXU(theta)
__device__ void enc_super(float th, cplx S[4][4]) {
  cplx T[4][4];
  rxu_super(th, S);
  dep_super(EPC_1Q, T); mat4_lmul(T, S);
  ad_super(1.f - expf(-T_RX / T1C), T); mat4_lmul(T, S);
  pd_super(1.f - expf(-T_RX / T2C), T); mat4_lmul(T, S);
}
// rot_block superop: DEP1*Drz(a) * [noisy_rx(th)] * DEP1*Drz(phi)
__device__ void rot_super(const float* w3, cplx S[4][4]) {
  float th = w3[0], phi = w3[1], al = w3[2];
  cplx T[4][4];
  // start: identity, then rz(phi)
  for (int m = 0; m < 4; ++m)
    for (int k = 0; k < 4; ++k) S[m][k] = {(m == k) ? 1.f : 0.f, 0.f};
  rz_lmul(phi, S);
  dep_super(EPC_1Q, T); mat4_lmul(T, S);
  rxu_super(th, T); mat4_lmul(T, S);
  dep_super(EPC_1Q, T); mat4_lmul(T, S);
  ad_super(1.f - expf(-T_RX / T1C), T); mat4_lmul(T, S);
  pd_super(1.f - expf(-T_RX / T2C), T); mat4_lmul(T, S);
  rz_lmul(al, S);
  dep_super(EPC_1Q, T); mat4_lmul(T, S);
}
// per-qubit rzz noise superop: PD_ZZ * AD_ZZ * DEP2
__device__ void rzz_noise_super(cplx S[4][4]) {
  cplx T[4][4];
  dep_super(EPC_2Q, S);
  ad_super(1.f - expf(-T_RZZ / T1C), T); mat4_lmul(T, S);
  pd_super(1.f - expf(-T_RZZ / T2C), T); mat4_lmul(T, S);
}

// Kron of two 4x4 superops into 16x16 real/imag planes (row-major, 16 floats per row)
__device__ void kron_store(float* dre, float* dim, const cplx A[4][4], const cplx B[4][4]) {
  for (int mA = 0; mA < 4; ++mA)
    for (int kA = 0; kA < 4; ++kA)
      for (int mB = 0; mB < 4; ++mB)
        for (int kB = 0; kB < 4; ++kB) {
          cplx v = cmul(A[mA][kA], B[mB][kB]);
          int m = mA * 4 + mB, k = kA * 4 + kB;
          dre[m * 16 + k] = v.re;
          dim[m * 16 + k] = v.im;
        }
}

// ---------------- prep kernel: 13 batch-independent pair superoperators ----------------
// layout: sops[idx][0][256]=Re, sops[idx][1][256]=Im   (idx*512 floats)
// idx 0..7 : rot pair superops, layer l=idx>>2, pair p=idx&3 (qubits 2p,2p+1)
// idx 8    : rzz noise pair superop (same for every rzz position)
// idx 9..12: final rot pair superops
__global__ void build_superops(const float* __restrict__ qw, float* __restrict__ sops) {
  int tid = threadIdx.x;
  if (tid >= NSOPS) return;
  cplx SA[4][4], SB[4][4];
  if (tid < 8) {
    int l = tid >> 2, p = tid & 3, q0 = 2 * p;
    rot_super(qw + (l * NQ + q0) * 3, SA);
    rot_super(qw + (l * NQ + q0 + 1) * 3, SB);
  } else if (tid == 8) {
    rzz_noise_super(SA);
    for (int m = 0; m < 4; ++m) for (int k = 0; k < 4; ++k) SB[m][k] = SA[m][k];
  } else {
    int p = tid - 9, q0 = 2 * p;
    const float* fin = qw + 62;  // SW+SE = 48+14
    rot_super(fin + q0 * 3, SA);
    rot_super(fin + (q0 + 1) * 3, SB);
  }
  kron_store(sops + tid * 512, sops + tid * 512 + 256, SA, SB);
}

// ---------------- index helpers ----------------
__device__ inline int ins2(int x, int b0, int b1, int s) {
  int hi = x >> s, lo = x & ((1 << s) - 1);
  return (hi << (s + 2)) | (b0 << (s + 1)) | (b1 << s) | lo;
}
__device__ inline int addr_of(int m, int rrest, int crest, int s) {
  int i0 = (m >> 3) & 1, j0 = (m >> 2) & 1, i1 = (m >> 1) & 1, j1 = m & 1;
  return (ins2(rrest, i0, i1, s) << 8) | ins2(crest, j0, j1, s);
}

// ---------------- WMMA pair-superoperator application ----------------
// rho' = S * R  where R[m, t] = rho[row(m_bits,t), col(m_bits,t)], m in [0,16), t in [0,4096).
// Complex via 4 real 16x16x(16) matmuls, each as 4 chained v_wmma_f32_16x16x4_f32.
// Column groups of 16 partition rho -> in-place per wave is safe; waves own disjoint tiles.
__device__ void apply_pair(float* __restrict__ RE, float* __restrict__ IM,
                           const float* Sr, const float* Si,
                           int q, int lane, int wave) {
  const int s = 6 - q;            // bits below the (q,q+1) pair (qubit0 = MSB)
  const int M = lane & 15, kh = lane >> 4;
  // A fragments: 16x4 f32 layout -> lane M=lane&15, VGPR0/1 = K = (lane>>4)*2 (+1)
  v2f Ar[4], Ai[4], An[4];
#pragma unroll
  for (int kc = 0; kc < 4; ++kc) {
    int col = kc * 4 + kh * 2;
    v2f ar = {Sr[M * 16 + col], Sr[M * 16 + col + 1]};
    v2f ai = {Si[M * 16 + col], Si[M * 16 + col + 1]};
    Ar[kc] = ar; Ai[kc] = ai; An[kc] = -ai;
  }
  for (int t = wave; t < 256; t += 8) {
    int tcol = t * 16 + (lane & 15);
    int rrest = tcol >> 6, crest = tcol & 63;
    // B fragments: 4x16 f32 layout -> lane n=lane&15 col, VGPR0/1 = rows (lane>>4)*2 (+1)
    v2f Br[4], Bi[4];
#pragma unroll
    for (int kc = 0; kc < 4; ++kc) {
      int m0 = kc * 4 + kh * 2;
      int a0 = addr_of(m0, rrest, crest, s);
      int a1 = addr_of(m0 + 1, rrest, crest, s);
      v2f br = {RE[a0], RE[a1]};
      v2f bi = {IM[a0], IM[a1]};
      Br[kc] = br; Bi[kc] = bi;
    }
    if (t + 8 < 256) {  // prefetch next tile's first cacheline
      int nt = (t + 8) * 16 + (lane & 15);
      __builtin_prefetch(&RE[addr_of(kh * 2, nt >> 6, nt & 63, s)], 0, 1);
    }
    v8f cr = {};
    v8f ci = {};
    // Pr = Sr*Rr - Si*Ri ; Pi = Sr*Ri + Si*Rr
#pragma unroll
    for (int kc = 0; kc < 4; ++kc)
      cr = __builtin_amdgcn_wmma_f32_16x16x4_f32(false, Ar[kc], false, Br[kc], (short)0, cr, false, false);
#pragma unroll
    for (int kc = 0; kc < 4; ++kc)
      cr = __builtin_amdgcn_wmma_f32_16x16x4_f32(false, An[kc], false, Bi[kc], (short)0, cr, false, false);
#pragma unroll
    for (int kc = 0; kc < 4; ++kc)
      ci = __builtin_amdgcn_wmma_f32_16x16x4_f32(false, Ar[kc], false, Bi[kc], (short)0, ci, false, false);
#pragma unroll
    for (int kc = 0; kc < 4; ++kc)
      ci = __builtin_amdgcn_wmma_f32_16x16x4_f32(false, Ai[kc], false, Br[kc], (short)0, ci, false, false);
    // D layout: lane n=lane&15 col, VGPR v -> output row v + 8*(lane>>4)
#pragma unroll
    for (int v = 0; v < 8; ++v) {
      int a = addr_of(v + kh * 8, rrest, crest, s);
      RE[a] = cr[v];
      IM[a] = ci[v];
    }
  }
}

// elementwise rzz(theta) phase on (q,q+1): rho[r,c] *= exp(-i*theta/2*(s_r - s_c))
__device__ void diag_rzz(float* RE, float* IM, float th, int q, int tid) {
  for (int i = tid; i < NELEM; i += 256) {
    int r = i >> 8, c = i & 255;
    int br = ((r >> (7 - q)) ^ (r >> (6 - q))) & 1;  // 1 -> s_r = -1
    int bc = ((c >> (7 - q)) ^ (c >> (6 - q))) & 1;
    if (br == bc) continue;                          // net phase 1
    float sr = br ? -1.f : 1.f;
    float sn, cs;
    __sincosf(-th * sr, &sn, &cs);
    float re = RE[i], im = IM[i];
    RE[i] = re * cs - im * sn;
    IM[i] = re * sn + im * cs;
  }
}

#define BSYNC() do { __threadfence_block(); __syncthreads(); } while (0)

// ---------------- main kernel: one workgroup (8 wave32) per batch element ----------------
__global__ void __launch_bounds__(256)
simulate_kernel(const float* __restrict__ x, const float* __restrict__ fmin,
                const float* __restrict__ qw, const float* __restrict__ dk,
                const float* __restrict__ db, const float* __restrict__ sops,
                float* __restrict__ rho_ws, float* __restrict__ out) {
  int b = blockIdx.x;
  int tid = threadIdx.x, lane = tid & 31, wave = tid >> 5;
  float* RE = rho_ws + (size_t)b * 2u * NELEM;
  float* IM = RE + NELEM;

  __shared__ float sS[NSOPS + 4][2][256];  // 13 precomputed + 4 per-batch encoding
  __shared__ float red[DIM];

  // stage batch-independent superops into LDS
  for (int i = tid; i < NSOPS * 512; i += 256) (&sS[0][0][0])[i] = sops[i];

  // per-batch encoding pair superops: noisy_rx(x[b][q]*fmin) on (2p, 2p+1)
  if (tid < 4) {
    int q0 = 2 * tid;
    cplx SA[4][4], SB[4][4];
    enc_super(x[b * NQ + q0] * fmin[0], SA);
    enc_super(x[b * NQ + q0 + 1] * fmin[0], SB);
    kron_store(sS[NSOPS + tid][0], sS[NSOPS + tid][1], SA, SB);
  }

  // rho = |0><0|
  for (int i = tid; i < NELEM; i += 256) {
    RE[i] = (i == 0) ? 1.f : 0.f;
    IM[i] = 0.f;
  }
  BSYNC();

  // encoding layer
  for (int p = 0; p < 4; ++p) {
    apply_pair(RE, IM, sS[NSOPS + p][0], sS[NSOPS + p][1], 2 * p, lane, wave);
    BSYNC();
  }
  // variational layers
  for (int l = 0; l < LAYERS; ++l) {
    for (int p = 0; p < 4; ++p) {
      apply_pair(RE, IM, sS[l * 4 + p][0], sS[l * 4 + p][1], 2 * p, lane, wave);
      BSYNC();
    }
    for (int q = 0; q < NQ - 1; ++q) {
      diag_rzz(RE, IM, qw[48 + l * (NQ - 1) + q], q, tid);
      BSYNC();
      apply_pair(RE, IM, sS[8][0], sS[8][1], q, lane, wave);
      BSYNC();
    }
  }
  // final rotations
  for (int p = 0; p < 4; ++p) {
    apply_pair(RE, IM, sS[9 + p][0], sS[9 + p][1], 2 * p, lane, wave);
    BSYNC();
  }

  // <Z0> = sum_r (+/-1) * Re rho[r,r]  (qubit0 = MSB)
  float v = RE[tid * 256 + tid];
  red[tid] = (tid & 0x80) ? -v : v;
  __syncthreads();
  if (tid == 0) {
    float ssum = 0.f;
    for (int i = 0; i < DIM; ++i) ssum += red[i];
    out[b] = ssum * dk[0] + db[0];
  }
}

// ---------------- launch ----------------
extern "C" void kernel_launch(void* const* d_in, const int* in_sizes, int n_in,
                              void* d_out, int out_size, void* d_ws, size_t ws_size,
                              hipStream_t stream) {
  const float* x    = (const float*)d_in[0];  // (64, 8)
  const float* fmin = (const float*)d_in[1];  // (1,)
  const float* qw   = (const float*)d_in[2];  // (86,)
  const float* dk   = (const float*)d_in[3];  // (1,1)
  const float* db   = (const float*)d_in[4];  // (1,)
  float* out = (float*)d_out;                 // (64,1)

  float* sops = (float*)d_ws;                 // 13*512 floats
  float* rho  = sops + NSOPS * 512;           // 64 * 2 * 65536 floats (32 MB, L2-resident)

  build_superops<<<1, 32, 0, stream>>>(qw, sops);
  simulate_kernel<<<BATCH, 256, 0, stream>>>(x, fmin, qw, dk, db, sops, rho, out);
}